// TransitionDown_90580860272884
// MI455X (gfx1250) — compile-verified
//
#include <hip/hip_runtime.h>

// ---------------------------------------------------------------------------
// TransitionDown for MI455X (gfx1250, wave32).
//   n = 32768 points, cin = 128, cout = 256, m = n/4 = 8192, k = 16.
// Output: y (8192x256) then p2 (8192x3), concatenated flat, fp32.
// ---------------------------------------------------------------------------

typedef __attribute__((ext_vector_type(2))) float v2f;
typedef __attribute__((ext_vector_type(8))) float v8f;

#define N_PTS   32768
#define M_PTS   8192
#define CIN     128
#define COUT    256
#define KNN_K   16
#define FLTMAX  3.402823466e+38f

// ---------------------------------------------------------------------------
// 1) Furthest point sampling: single 1024-thread workgroup, 8191 serial steps.
//    min_d lives in registers (32 per thread); argmax via wave32 shuffle
//    reduction + 32-partial LDS reduction. Writes sampled indices and p2.
// ---------------------------------------------------------------------------
__global__ __launch_bounds__(1024) void fps_kernel(const float* __restrict__ p1,
                                                   int* __restrict__ sampled,
                                                   float* __restrict__ p2) {
  const int tid = threadIdx.x;
  __shared__ float redv[32];
  __shared__ int   redi[32];
  __shared__ int   s_last;

  float mind[32];
#pragma unroll
  for (int i = 0; i < 32; ++i) mind[i] = FLTMAX;

  if (tid == 0) {
    sampled[0] = 0;
    p2[0] = p1[0]; p2[1] = p1[1]; p2[2] = p1[2];
    s_last = 0;
  }
  __syncthreads();
  int last = 0;

  for (int step = 1; step < M_PTS; ++step) {
    const float lx = p1[last * 3 + 0];
    const float ly = p1[last * 3 + 1];
    const float lz = p1[last * 3 + 2];

    float best = -1.0f;
    int   bidx = 0x7fffffff;
#pragma unroll
    for (int i = 0; i < 32; ++i) {
      const int j = tid + (i << 10);
      const float dx = p1[j * 3 + 0] - lx;
      const float dy = p1[j * 3 + 1] - ly;
      const float dz = p1[j * 3 + 2] - lz;
      const float d  = dx * dx + dy * dy + dz * dz;
      const float nm = fminf(mind[i], d);
      mind[i] = nm;
      if (nm > best) { best = nm; bidx = j; }   // strided j increasing -> lowest idx on tie
    }
    // wave32 reduction (argmax, tie -> lowest index, matching jnp.argmax)
#pragma unroll
    for (int off = 16; off > 0; off >>= 1) {
      const float ov = __shfl_xor(best, off, 32);
      const int   oi = __shfl_xor(bidx, off, 32);
      if (ov > best || (ov == best && oi < bidx)) { best = ov; bidx = oi; }
    }
    if ((tid & 31) == 0) { redv[tid >> 5] = best; redi[tid >> 5] = bidx; }
    __syncthreads();
    if (tid < 32) {
      best = redv[tid]; bidx = redi[tid];
#pragma unroll
      for (int off = 16; off > 0; off >>= 1) {
        const float ov = __shfl_xor(best, off, 32);
        const int   oi = __shfl_xor(bidx, off, 32);
        if (ov > best || (ov == best && oi < bidx)) { best = ov; bidx = oi; }
      }
      if (tid == 0) {
        s_last = bidx;
        sampled[step] = bidx;
        p2[step * 3 + 0] = p1[bidx * 3 + 0];
        p2[step * 3 + 1] = p1[bidx * 3 + 1];
        p2[step * 3 + 2] = p1[bidx * 3 + 2];
      }
    }
    __syncthreads();
    last = s_last;
  }
}

// ---------------------------------------------------------------------------
// 2) h = x @ W + b via V_WMMA_F32_16X16X4_F32. One wave per 16x16 tile of h.
//    A (16x4 f32): lane l<16 holds row l, K={0,1}; lane l+16 holds K={2,3}.
//    B (4x16 f32): lane l<16 holds col l, K={0,1}; lane l+16 holds K={2,3}.
//    C/D: VGPR r, lane l -> row r + 8*(l>>4), col l&15.
// ---------------------------------------------------------------------------
__global__ __launch_bounds__(256) void gemm_bias_kernel(const float* __restrict__ x,
                                                        const float* __restrict__ W,
                                                        const float* __restrict__ bias,
                                                        float* __restrict__ h) {
  const int lane = threadIdx.x & 31;
  const int wave = threadIdx.x >> 5;
  const int l15  = lane & 15;
  const int hi   = lane >> 4;
  const int mBase = blockIdx.x * 16;
  const int nBase = (blockIdx.y * 8 + wave) * 16;

  const float* arow = x + (size_t)(mBase + l15) * CIN + hi * 2;
  const float* bcol = W + (size_t)(hi * 2) * COUT + nBase + l15;

  v8f acc = {0.f, 0.f, 0.f, 0.f, 0.f, 0.f, 0.f, 0.f};
#pragma unroll
  for (int kk = 0; kk < CIN / 4; ++kk) {
    v2f a = *(const v2f*)(arow + kk * 4);          // contiguous K pair, 8B aligned
    v2f b;
    b.x = bcol[(size_t)kk * 4 * COUT];
    b.y = bcol[(size_t)kk * 4 * COUT + COUT];
    acc = __builtin_amdgcn_wmma_f32_16x16x4_f32(false, a, false, b, (short)0, acc,
                                                false, false);
  }
  const int col = nBase + l15;
  const float bv = bias[col];
#pragma unroll
  for (int r = 0; r < 8; ++r) {
    const int row = mBase + r + 8 * hi;
    h[(size_t)row * COUT + col] = acc[r] + bv;
  }
}

// ---------------------------------------------------------------------------
// 3) BN partial sums (deterministic tree, no float atomics): each block owns
//    128 rows, thread c accumulates column c (coalesced).
// ---------------------------------------------------------------------------
__global__ __launch_bounds__(256) void bnstat_partial(const float* __restrict__ h,
                                                      float* __restrict__ psum,
                                                      float* __restrict__ psq) {
  const int c  = threadIdx.x;
  const int rb = blockIdx.x * 128;
  float s = 0.f, ss = 0.f;
  for (int r = 0; r < 128; ++r) {
    const float v = h[(size_t)(rb + r) * COUT + c];
    s += v;
    ss += v * v;
  }
  psum[blockIdx.x * COUT + c] = s;
  psq[blockIdx.x * COUT + c]  = ss;
}

__global__ __launch_bounds__(256) void bnstat_final(const float* __restrict__ psum,
                                                    const float* __restrict__ psq,
                                                    const float* __restrict__ gamma,
                                                    const float* __restrict__ beta,
                                                    float* __restrict__ scale,
                                                    float* __restrict__ shift) {
  const int c = threadIdx.x;
  float s = 0.f, ss = 0.f;
  for (int bkt = 0; bkt < N_PTS / 128; ++bkt) {
    s += psum[bkt * COUT + c];
    ss += psq[bkt * COUT + c];
  }
  const float inv_n = 1.0f / (float)N_PTS;
  const float mu   = s * inv_n;
  const float var  = fmaxf(ss * inv_n - mu * mu, 0.f);
  const float rstd = rsqrtf(var + 1e-5f);
  const float sc   = rstd * gamma[c];
  scale[c] = sc;
  shift[c] = beta[c] - mu * sc;
}

// ---------------------------------------------------------------------------
// 4) kNN: one thread per query, p1 tiled through LDS (2048 pts = 24 KB),
//    fully-unrolled 16-deep register insertion sort with early reject.
// ---------------------------------------------------------------------------
#define KTILE 2048
__global__ __launch_bounds__(128) void knn_kernel(const float* __restrict__ p1,
                                                  const int* __restrict__ sampled,
                                                  int* __restrict__ nbrs) {
  __shared__ float sx[KTILE], sy[KTILE], sz[KTILE];
  const int q    = blockIdx.x * 128 + threadIdx.x;
  const int self = sampled[q];
  const float qx = p1[self * 3 + 0];
  const float qy = p1[self * 3 + 1];
  const float qz = p1[self * 3 + 2];

  float bd[KNN_K];
  int   bi[KNN_K];
#pragma unroll
  for (int s = 0; s < KNN_K; ++s) { bd[s] = FLTMAX; bi[s] = -1; }

  for (int base = 0; base < N_PTS; base += KTILE) {
    __syncthreads();
    for (int t = threadIdx.x; t < KTILE; t += 128) {
      const int j = base + t;
      sx[t] = p1[j * 3 + 0];
      sy[t] = p1[j * 3 + 1];
      sz[t] = p1[j * 3 + 2];
    }
    __syncthreads();
    for (int t = 0; t < KTILE; ++t) {
      const int j = base + t;
      const float dx = sx[t] - qx, dy = sy[t] - qy, dz = sz[t] - qz;
      const float d  = dx * dx + dy * dy + dz * dz;
      if (d < bd[KNN_K - 1] && j != self) {
        float dc = d; int ic = j;
#pragma unroll
        for (int s = 0; s < KNN_K; ++s) {
          if (dc < bd[s]) {
            float tf = bd[s]; bd[s] = dc; dc = tf;
            int   ti = bi[s]; bi[s] = ic; ic = ti;
          }
        }
      }
    }
  }
#pragma unroll
  for (int s = 0; s < KNN_K; ++s) nbrs[q * KNN_K + s] = bi[s];
}

// ---------------------------------------------------------------------------
// 5) Fused BN-affine + ReLU + neighbor max-pool. One block per query; thread c
//    owns output channel c (coalesced row reads of h).
// ---------------------------------------------------------------------------
__global__ __launch_bounds__(256) void gather_max_kernel(const float* __restrict__ h,
                                                         const int* __restrict__ nbrs,
                                                         const float* __restrict__ scale,
                                                         const float* __restrict__ shift,
                                                         float* __restrict__ y) {
  __shared__ int nb[KNN_K];
  const int q = blockIdx.x;
  const int c = threadIdx.x;
  if (c < KNN_K) nb[c] = nbrs[q * KNN_K + c];
  __syncthreads();
  const float sc = scale[c];
  const float sh = shift[c];
  float mx = 0.0f;  // relu(v) >= 0, so max over relu'd values starts at 0
#pragma unroll
  for (int t = 0; t < KNN_K; ++t) {
    const float v = fmaf(h[(size_t)nb[t] * COUT + c], sc, sh);
    mx = fmaxf(mx, v);
  }
  y[(size_t)q * COUT + c] = mx;
}

// ---------------------------------------------------------------------------
// Launcher
// ---------------------------------------------------------------------------
extern "C" void kernel_launch(void* const* d_in, const int* in_sizes, int n_in,
                              void* d_out, int out_size, void* d_ws, size_t ws_size,
                              hipStream_t stream) {
  const float* x     = (const float*)d_in[0];
  const float* p1    = (const float*)d_in[1];
  const float* W     = (const float*)d_in[2];
  const float* b     = (const float*)d_in[3];
  const float* gamma = (const float*)d_in[4];
  const float* beta  = (const float*)d_in[5];

  float* y  = (float*)d_out;                     // (8192, 256)
  float* p2 = y + (size_t)M_PTS * COUT;          // (8192, 3)

  char* w = (char*)d_ws;
  int*   sampled = (int*)w;    w += (size_t)M_PTS * sizeof(int);
  int*   nbrs    = (int*)w;    w += (size_t)M_PTS * KNN_K * sizeof(int);
  float* h       = (float*)w;  w += (size_t)N_PTS * COUT * sizeof(float);
  float* psum    = (float*)w;  w += (size_t)(N_PTS / 128) * COUT * sizeof(float);
  float* psq     = (float*)w;  w += (size_t)(N_PTS / 128) * COUT * sizeof(float);
  float* scale   = (float*)w;  w += (size_t)COUT * sizeof(float);
  float* shift   = (float*)w;  w += (size_t)COUT * sizeof(float);

  fps_kernel<<<1, 1024, 0, stream>>>(p1, sampled, p2);
  gemm_bias_kernel<<<dim3(N_PTS / 16, 2), 256, 0, stream>>>(x, W, b, h);
  bnstat_partial<<<N_PTS / 128, 256, 0, stream>>>(h, psum, psq);
  bnstat_final<<<1, 256, 0, stream>>>(psum, psq, gamma, beta, scale, shift);
  knn_kernel<<<M_PTS / 128, 128, 0, stream>>>(p1, sampled, nbrs);
  gather_max_kernel<<<M_PTS, 256, 0, stream>>>(h, nbrs, scale, shift, y);
}